// ConfidenceLabelPropagation_27178553049106
// MI455X (gfx1250) — compile-verified
//
#include <hip/hip_runtime.h>
#include <stdint.h>

#define EPSF 1e-8f
#define INV_LOG64 0.24044917348149793f  // 1/ln(64)

// ---------------------------------------------------------------- helpers ---
__device__ __forceinline__ float clampf(float x, float a, float b) {
  return fminf(fmaxf(x, a), b);
}
__device__ __forceinline__ float sigmoidf(float x) {
  return 1.0f / (1.0f + expf(-x));
}
__device__ __forceinline__ float wave_sum(float v) {
#pragma unroll
  for (int m = 16; m > 0; m >>= 1) v += __shfl_xor(v, m, 32);
  return v;
}
// top-2 across the wave; inputs are per-lane (hi, lo) candidates
__device__ __forceinline__ void wave_top2(float& hi, float& lo) {
#pragma unroll
  for (int m = 16; m > 0; m >>= 1) {
    float h = __shfl_xor(hi, m, 32);
    float l = __shfl_xor(lo, m, 32);
    float nh = fmaxf(hi, h);
    float nl = fmaxf(fminf(hi, h), fmaxf(lo, l));
    hi = nh; lo = nl;
  }
}
// guaranteed-native no-return f32 scatter atomic, device scope (L2 atomic units)
__device__ __forceinline__ void gatomic_add(float* p, float v) {
  asm volatile("global_atomic_add_f32 %0, %1, off scope:SCOPE_DEV"
               :: "v"(p), "v"(v) : "memory");
}

// CDNA5 async global->LDS copy (ASYNCcnt-tracked), per-lane 4B
#define ASYNC_LDS_B32(lds_off, gptr)                                   \
  asm volatile("global_load_async_to_lds_b32 %0, %1, off"              \
               :: "v"(lds_off), "v"(gptr) : "memory")
#define WAIT_ASYNC_3() asm volatile("s_wait_asynccnt 3" ::: "memory")
#define WAIT_ASYNC_0() asm volatile("s_wait_asynccnt 0" ::: "memory")

// ---------------------------------------------------------------- kernels ---

// zero the small scalar / prior area of the workspace
__global__ void k_init(float* sc) {
  if (threadIdx.x < 256) sc[threadIdx.x] = 0.0f;
}

// grid-stride zero fill
__global__ __launch_bounds__(256) void k_zero(float* __restrict__ p, long n) {
  long i = (long)blockIdx.x * blockDim.x + threadIdx.x;
  long s = (long)gridDim.x * blockDim.x;
  for (; i < n; i += s) p[i] = 0.0f;
}

// seed = relu(logits) -> prop (d_out); accumulate sum(mass(seed)), sum(clustering)
__global__ __launch_bounds__(256) void k_seed_init(
    const float* __restrict__ logits, const float* __restrict__ sf,
    float* __restrict__ prop, float* __restrict__ sc, int N) {
  const int wv = threadIdx.x >> 5, lane = threadIdx.x & 31;
  const int wpb = blockDim.x >> 5;
  int n = blockIdx.x * wpb + wv;
  const int stride = gridDim.x * wpb;
  float accM = 0.0f, accCl = 0.0f;
  for (; n < N; n += stride) {
    const size_t ro = ((size_t)n << 6) + 2 * lane;
    float sx = fmaxf(logits[ro], 0.0f);
    float sy = fmaxf(logits[ro + 1], 0.0f);
    prop[ro] = sx; prop[ro + 1] = sy;
    float mass = wave_sum(sx + sy);
    if (lane == 0) { accM += mass; accCl += sf[2 * n + 1]; }
  }
  if (lane == 0) { gatomic_add(&sc[0], accM); gatomic_add(&sc[2], accCl); }
}

// confidence(seed), source_gate; accumulate sum(conf), sum(conf*seed) per class
__global__ __launch_bounds__(256) void k_conf(
    const float* __restrict__ logits, float* __restrict__ confv,
    float* __restrict__ sgate, float* __restrict__ sc,
    float* __restrict__ prior_acc, int N) {
  const int wv = threadIdx.x >> 5, lane = threadIdx.x & 31;
  const int wpb = blockDim.x >> 5;
  int n = blockIdx.x * wpb + wv;
  const int stride = gridDim.x * wpb;
  const float rscale = 1.0f / fmaxf(sc[0] / (float)N, EPSF);
  float accC = 0.0f, aPx = 0.0f, aPy = 0.0f;
  for (; n < N; n += stride) {
    const size_t ro = ((size_t)n << 6) + 2 * lane;
    float sx = fmaxf(logits[ro], 0.0f);
    float sy = fmaxf(logits[ro + 1], 0.0f);
    float mass = wave_sum(sx + sy);
    float inv = 1.0f / (mass + EPSF);
    float nx = sx * inv, ny = sy * inv;
    float ent = wave_sum(-(nx * logf(nx + EPSF) + ny * logf(ny + EPSF)));
    float cert = 1.0f - ent * INV_LOG64;
    float mag = tanhf(mass * rscale);
    float conf = clampf(0.5f * cert + 0.5f * mag, 0.0f, 1.0f);
    if (lane == 0) {
      confv[n] = conf;
      sgate[n] = sigmoidf(8.0f * (conf - 0.55f));
      accC += conf;
    }
    aPx += conf * sx; aPy += conf * sy;
  }
  if (lane == 0) gatomic_add(&sc[1], accC);
  gatomic_add(prior_acc + 2 * lane, aPx);
  gatomic_add(prior_acc + 2 * lane + 1, aPy);
}

// global_prior[c], graph_scale
__global__ void k_finalize(float* sc, const float* prior_acc, float* prior, int N) {
  int c = threadIdx.x;
  if (c < 64) prior[c] = prior_acc[c] / fmaxf(sc[1], EPSF);
  if (c == 0) sc[3] = clampf(1.0f - sc[2] / (float)N, 0.2f, 1.0f);
}

// loop-invariant edge precompute: ecoef[e] = w*gate[src]; lden[dst] += ecoef
__global__ __launch_bounds__(256) void k_edge_pre(
    const int* __restrict__ esrc, const int* __restrict__ edst,
    const float* __restrict__ ew, const float* __restrict__ sgate,
    float* __restrict__ ecoef, float* __restrict__ lden, int E) {
  int e = blockIdx.x * blockDim.x + threadIdx.x;
  const int stride = gridDim.x * blockDim.x;
  for (; e < E; e += stride) {
    float c = ew[e] * sgate[esrc[e]];
    ecoef[e] = c;
    gatomic_add(lden + edst[e], c);
  }
}

// SpMM step: local_num[dst] += ecoef[e] * prop[src]
// edge triples staged into LDS via async copies, double-buffered per wave
__global__ __launch_bounds__(256) void k_edge_prop(
    const int* __restrict__ esrc, const int* __restrict__ edst,
    const float* __restrict__ ecoef, const float* __restrict__ prop,
    float* __restrict__ lnum, int E) {
  __shared__ uint32_t sh_s[8][2][32];
  __shared__ uint32_t sh_d[8][2][32];
  __shared__ float    sh_c[8][2][32];
  const int wv = threadIdx.x >> 5, lane = threadIdx.x & 31;
  const int nCh = (E + 31) >> 5;
  int ch = blockIdx.x * 8 + wv;
  const int stride = gridDim.x * 8;
  if (ch >= nCh) return;
  uint32_t lsS[2], lsD[2], lsC[2];
  lsS[0] = (uint32_t)(uintptr_t)&sh_s[wv][0][lane];
  lsS[1] = (uint32_t)(uintptr_t)&sh_s[wv][1][lane];
  lsD[0] = (uint32_t)(uintptr_t)&sh_d[wv][0][lane];
  lsD[1] = (uint32_t)(uintptr_t)&sh_d[wv][1][lane];
  lsC[0] = (uint32_t)(uintptr_t)&sh_c[wv][0][lane];
  lsC[1] = (uint32_t)(uintptr_t)&sh_c[wv][1][lane];
  {
    int idx = ch * 32 + lane; idx = idx < E ? idx : (E - 1);
    ASYNC_LDS_B32(lsS[0], (const void*)(esrc + idx));
    ASYNC_LDS_B32(lsD[0], (const void*)(edst + idx));
    ASYNC_LDS_B32(lsC[0], (const void*)(ecoef + idx));
  }
  int p = 0;
  for (; ch < nCh; ch += stride) {
    int nxt = ch + stride;
    if (nxt < nCh) {
      int idx = nxt * 32 + lane; idx = idx < E ? idx : (E - 1);
      ASYNC_LDS_B32(lsS[p ^ 1], (const void*)(esrc + idx));
      ASYNC_LDS_B32(lsD[p ^ 1], (const void*)(edst + idx));
      ASYNC_LDS_B32(lsC[p ^ 1], (const void*)(ecoef + idx));
      WAIT_ASYNC_3();
    } else {
      WAIT_ASYNC_0();
    }
    int base = ch * 32;
    int cnt = E - base; cnt = cnt < 32 ? cnt : 32;
    uint32_t se = sh_s[wv][p][lane];
    uint32_t de = sh_d[wv][p][lane];
    float ce = sh_c[wv][p][lane];
    // warm caches for the 32 gathered rows before the broadcast loop
    __builtin_prefetch((const void*)(prop + ((size_t)se << 6)), 0, 3);
    for (int j = 0; j < cnt; ++j) {
      int sj = __shfl((int)se, j, 32);
      int dj = __shfl((int)de, j, 32);
      float cj = __shfl(ce, j, 32);
      float2 sv = *(const float2*)(prop + ((size_t)sj << 6) + 2 * lane);
      float* drow = lnum + ((size_t)dj << 6) + 2 * lane;
      gatomic_add(drow, cj * sv.x);
      gatomic_add(drow + 1, cj * sv.y);
    }
    p ^= 1;
  }
}

// per-node propagation update (one wave per node, 2 classes/lane)
__global__ __launch_bounds__(256) void k_node_update(
    const float* __restrict__ lnum, const float* __restrict__ lden,
    float* __restrict__ prop, const float* __restrict__ logits,
    const float* __restrict__ sf, const float* __restrict__ confv,
    const float* __restrict__ prior, const float* __restrict__ sc, int N) {
  const int wv = threadIdx.x >> 5, lane = threadIdx.x & 31;
  const int wpb = blockDim.x >> 5;
  int n = blockIdx.x * wpb + wv;
  const int stride = gridDim.x * wpb;
  const float gscale = sc[3];
  const float px = prior[2 * lane], py = prior[2 * lane + 1];
  for (; n < N; n += stride) {
    const size_t ro = ((size_t)n << 6) + 2 * lane;
    float den = fmaxf(lden[n], EPSF);
    float rinv = 1.0f / den;
    float2 nm = *(const float2*)(lnum + ro);
    float fx = 0.95f * (nm.x * rinv) + 0.05f * px;
    float fy = 0.95f * (nm.y * rinv) + 0.05f * py;
    float2 pv = *(const float2*)(prop + ro);
    float2 lg = *(const float2*)(logits + ro);
    float sx = fmaxf(lg.x, 0.0f), sy = fmaxf(lg.y, 0.0f);
    float dp  = wave_sum(pv.x * fx + pv.y * fy);
    float npr = wave_sum(pv.x * pv.x + pv.y * pv.y);
    float nf  = wave_sum(fx * fx + fy * fy);
    float dsd = wave_sum(sx * fx + sy * fy);
    float nsd = wave_sum(sx * sx + sy * sy);
    float agree  = clampf((dp  / (fmaxf(sqrtf(npr), EPSF) * fmaxf(sqrtf(nf), EPSF)) + 1.0f) * 0.5f, 0.0f, 1.0f);
    float sagree = clampf((dsd / (fmaxf(sqrtf(nsd), EPSF) * fmaxf(sqrtf(nf), EPSF)) + 1.0f) * 0.5f, 0.0f, 1.0f);
    float conf = confv[n];
    float anchor = clampf(0.6f + 0.2f * conf, 0.0f, 0.995f);
    float unc = 1.0f - conf;
    float rgate = sigmoidf(8.0f * (0.5f - conf));
    float ldeg = clampf(1.0f - sf[2 * n], 0.0f, 1.0f);
    float clu = sf[2 * n + 1];
    float lclu = clampf(1.0f - clu, 0.0f, 1.0f);
    float selective = clampf(unc + 0.25f * ldeg + 0.2f * sagree + 0.2f * lclu, 0.0f, 1.0f);
    float ug = rgate * selective * agree * (1.0f - anchor);
    float rs = 0.15f * gscale * ug;  // GRAPH_SCALE_BIAS = 1.0
    float nxv = anchor * sx + (1.0f - anchor) * pv.x + rs * (fx - pv.x);
    float nyv = anchor * sy + (1.0f - anchor) * pv.y + rs * (fy - pv.y);
    prop[ro]     = fmaxf(nxv, 0.0f);
    prop[ro + 1] = fmaxf(nyv, 0.0f);
  }
}

// final context SpMMs: base_ctx += w*relu(logits[src]); prop_ctx += w*prop[src]
__global__ __launch_bounds__(256) void k_edge_final(
    const int* __restrict__ esrc, const int* __restrict__ edst,
    const float* __restrict__ ew, const float* __restrict__ logits,
    const float* __restrict__ prop, float* __restrict__ bctx,
    float* __restrict__ pctx, int E) {
  __shared__ uint32_t sh_s[8][2][32];
  __shared__ uint32_t sh_d[8][2][32];
  __shared__ float    sh_w[8][2][32];
  const int wv = threadIdx.x >> 5, lane = threadIdx.x & 31;
  const int nCh = (E + 31) >> 5;
  int ch = blockIdx.x * 8 + wv;
  const int stride = gridDim.x * 8;
  if (ch >= nCh) return;
  uint32_t lsS[2], lsD[2], lsW[2];
  lsS[0] = (uint32_t)(uintptr_t)&sh_s[wv][0][lane];
  lsS[1] = (uint32_t)(uintptr_t)&sh_s[wv][1][lane];
  lsD[0] = (uint32_t)(uintptr_t)&sh_d[wv][0][lane];
  lsD[1] = (uint32_t)(uintptr_t)&sh_d[wv][1][lane];
  lsW[0] = (uint32_t)(uintptr_t)&sh_w[wv][0][lane];
  lsW[1] = (uint32_t)(uintptr_t)&sh_w[wv][1][lane];
  {
    int idx = ch * 32 + lane; idx = idx < E ? idx : (E - 1);
    ASYNC_LDS_B32(lsS[0], (const void*)(esrc + idx));
    ASYNC_LDS_B32(lsD[0], (const void*)(edst + idx));
    ASYNC_LDS_B32(lsW[0], (const void*)(ew + idx));
  }
  int p = 0;
  for (; ch < nCh; ch += stride) {
    int nxt = ch + stride;
    if (nxt < nCh) {
      int idx = nxt * 32 + lane; idx = idx < E ? idx : (E - 1);
      ASYNC_LDS_B32(lsS[p ^ 1], (const void*)(esrc + idx));
      ASYNC_LDS_B32(lsD[p ^ 1], (const void*)(edst + idx));
      ASYNC_LDS_B32(lsW[p ^ 1], (const void*)(ew + idx));
      WAIT_ASYNC_3();
    } else {
      WAIT_ASYNC_0();
    }
    int base = ch * 32;
    int cnt = E - base; cnt = cnt < 32 ? cnt : 32;
    uint32_t se = sh_s[wv][p][lane];
    uint32_t de = sh_d[wv][p][lane];
    float we = sh_w[wv][p][lane];
    __builtin_prefetch((const void*)(logits + ((size_t)se << 6)), 0, 3);
    __builtin_prefetch((const void*)(prop + ((size_t)se << 6)), 0, 3);
    for (int j = 0; j < cnt; ++j) {
      int sj = __shfl((int)se, j, 32);
      int dj = __shfl((int)de, j, 32);
      float wj = __shfl(we, j, 32);
      const size_t so = ((size_t)sj << 6) + 2 * lane;
      const size_t dofs = ((size_t)dj << 6) + 2 * lane;
      float2 lg = *(const float2*)(logits + so);
      float2 qv = *(const float2*)(prop + so);
      float ax = fmaxf(lg.x, 0.0f), ay = fmaxf(lg.y, 0.0f);
      gatomic_add(bctx + dofs, wj * ax);
      gatomic_add(bctx + dofs + 1, wj * ay);
      gatomic_add(pctx + dofs, wj * qv.x);
      gatomic_add(pctx + dofs + 1, wj * qv.y);
    }
    p ^= 1;
  }
}

// quality + accept blend -> out (in place over prop)
__global__ __launch_bounds__(256) void k_node_final(
    const float* __restrict__ logits, const float* __restrict__ prop,
    const float* __restrict__ bctx, const float* __restrict__ pctx,
    const float* __restrict__ sf, float* __restrict__ out, int N) {
  const int wv = threadIdx.x >> 5, lane = threadIdx.x & 31;
  const int wpb = blockDim.x >> 5;
  int n = blockIdx.x * wpb + wv;
  const int stride = gridDim.x * wpb;
  for (; n < N; n += stride) {
    const size_t ro = ((size_t)n << 6) + 2 * lane;
    float2 lg = *(const float2*)(logits + ro);
    float sx = fmaxf(lg.x, 0.0f), sy = fmaxf(lg.y, 0.0f);
    float2 pv = *(const float2*)(prop + ro);
    float2 bc = *(const float2*)(bctx + ro);
    float2 pc = *(const float2*)(pctx + ro);
    float db  = wave_sum(sx * bc.x + sy * bc.y);
    float nb  = wave_sum(sx * sx + sy * sy);
    float nbc = wave_sum(bc.x * bc.x + bc.y * bc.y);
    float dpz = wave_sum(pv.x * pc.x + pv.y * pc.y);
    float npz = wave_sum(pv.x * pv.x + pv.y * pv.y);
    float npc = wave_sum(pc.x * pc.x + pc.y * pc.y);
    float sb  = wave_sum(sx + sy);
    float sp  = wave_sum(pv.x + pv.y);
    float lqb = clampf((db  / (fmaxf(sqrtf(nb),  EPSF) * fmaxf(sqrtf(nbc), EPSF)) + 1.0f) * 0.5f, 0.0f, 1.0f);
    float lqp = clampf((dpz / (fmaxf(sqrtf(npz), EPSF) * fmaxf(sqrtf(npc), EPSF)) + 1.0f) * 0.5f, 0.0f, 1.0f);
    float ib = 1.0f / (sb + EPSF), ip = 1.0f / (sp + EPSF);
    float bh = fmaxf(sx, sy) * ib, bl = fminf(sx, sy) * ib;
    wave_top2(bh, bl);
    float ph = fmaxf(pv.x, pv.y) * ip, pl = fminf(pv.x, pv.y) * ip;
    wave_top2(ph, pl);
    float clu = sf[2 * n + 1];
    float qb = 0.7f * lqb + 0.2f * (bh - bl) + 0.1f * clu;
    float qp = 0.7f * lqp + 0.2f * (ph - pl) + 0.1f * clu;
    float acc = sigmoidf(12.0f * (qp - qb));
    out[ro]     = acc * pv.x + (1.0f - acc) * sx;
    out[ro + 1] = acc * pv.y + (1.0f - acc) * sy;
  }
}

// ------------------------------------------------------------------- host ---
extern "C" void kernel_launch(void* const* d_in, const int* in_sizes, int n_in,
                              void* d_out, int out_size, void* d_ws, size_t ws_size,
                              hipStream_t stream) {
  (void)n_in; (void)out_size; (void)ws_size;
  const int N = in_sizes[0] / 64;
  const int E = in_sizes[1];
  const float* logits = (const float*)d_in[0];
  const int* esrc = (const int*)d_in[1];
  const int* edst = (const int*)d_in[2];
  const float* ew = (const float*)d_in[3];
  const float* sf = (const float*)d_in[4];
  float* prop = (float*)d_out;  // d_out doubles as the propagated state

  // workspace layout (floats):
  // [0..15] scalars: 0=sum_mass_seed 1=sum_conf 2=sum_clust 3=graph_scale
  // [16..79] prior_acc[64], [80..143] prior[64], [256..) node/edge arrays
  float* sc = (float*)d_ws;
  float* prior_acc = sc + 16;
  float* prior = sc + 80;
  float* confv = sc + 256;
  float* sgate = confv + N;
  float* lden = sgate + N;
  float* lnum = lden + N;                 // N*64; reused as base_ctx at the end
  float* bctx = lnum;
  float* pctx = lnum + (size_t)N * 64;    // N*64
  float* ecoef = pctx + (size_t)N * 64;   // E

  k_init<<<1, 256, 0, stream>>>(sc);
  k_seed_init<<<1024, 256, 0, stream>>>(logits, sf, prop, sc, N);
  k_conf<<<1024, 256, 0, stream>>>(logits, confv, sgate, sc, prior_acc, N);
  k_finalize<<<1, 64, 0, stream>>>(sc, prior_acc, prior, N);

  // loop-invariant edge coefficients + denominators (computed once, used 3x)
  k_zero<<<1024, 256, 0, stream>>>(lden, (long)N);
  k_edge_pre<<<4096, 256, 0, stream>>>(esrc, edst, ew, sgate, ecoef, lden, E);

  for (int s = 0; s < 3; ++s) {
    k_zero<<<2048, 256, 0, stream>>>(lnum, (long)N * 64);
    k_edge_prop<<<4096, 256, 0, stream>>>(esrc, edst, ecoef, prop, lnum, E);
    k_node_update<<<2048, 256, 0, stream>>>(lnum, lden, prop, logits, sf, confv, prior, sc, N);
  }

  k_zero<<<4096, 256, 0, stream>>>(bctx, (long)N * 128);   // bctx + pctx contiguous
  k_edge_final<<<4096, 256, 0, stream>>>(esrc, edst, ew, logits, prop, bctx, pctx, E);
  k_node_final<<<2048, 256, 0, stream>>>(logits, prop, bctx, pctx, sf, prop, N);
}